// SparseTokenHandler_4904852652261
// MI455X (gfx1250) — compile-verified
//
#include <hip/hip_runtime.h>

typedef __attribute__((ext_vector_type(2))) float v2f;
typedef __attribute__((ext_vector_type(8))) float v8f;

#define B_    8
#define L_    8192
#define C_    512
#define NTOP  3687   // max(1, 4096 - 409)
#define NRAND 409    // int(4096 * 0.1)

// ---------------------------------------------------------------------------
// Kernel 1: per-token sum-of-squares via V_WMMA_F32_16X16X4_F32 with B = ones.
// Each wave handles 16 tokens; D[m][*] = sum_k x^2[m][k] after 128 steps.
// ---------------------------------------------------------------------------
__global__ __launch_bounds__(256) void norms_wmma_kernel(const float* __restrict__ x,
                                                         float* __restrict__ norms) {
    const int wave = threadIdx.x >> 5;
    const int lane = threadIdx.x & 31;
    const long tokBase = (long)blockIdx.x * 128 + (long)wave * 16;
    const int m  = lane & 15;        // A-matrix row (M) held by this lane
    const int kb = (lane >> 4) * 2;  // lanes 16-31 carry K=2,3 per ISA layout

    const float* rowp = x + (tokBase + m) * (long)C_ + kb;
    v8f acc = {};
    v2f bmat = {1.0f, 1.0f};         // B = ones -> row sums

    for (int c0 = 0; c0 < C_; c0 += 4) {
        float a0 = rowp[c0];
        float a1 = rowp[c0 + 1];
        v2f a = {a0 * a0, a1 * a1};
        acc = __builtin_amdgcn_wmma_f32_16x16x4_f32(
            /*neg_a=*/false, a, /*neg_b=*/false, bmat,
            /*c_mod=*/(short)0, acc, /*reuse_a=*/false, /*reuse_b=*/false);
    }
    // C/D layout: VGPR r -> M=r (lanes 0-15, N=lane), M=r+8 (lanes 16-31).
    if (lane == 0) {
        for (int r = 0; r < 8; ++r) norms[tokBase + r] = acc[r];
    } else if (lane == 16) {
        for (int r = 0; r < 8; ++r) norms[tokBase + 8 + r] = acc[r];
    }
}

// ---------------------------------------------------------------------------
// Threefry-2x32 (exact JAX schedule), key = (0, 42)
// ---------------------------------------------------------------------------
__device__ __forceinline__ unsigned rotl32(unsigned x, unsigned r) {
    return (x << r) | (x >> (32u - r));
}

__device__ void threefry2x32(unsigned k0, unsigned k1, unsigned c0, unsigned c1,
                             unsigned& o0, unsigned& o1) {
    unsigned ks2 = k0 ^ k1 ^ 0x1BD11BDAu;
    unsigned x0 = c0 + k0;
    unsigned x1 = c1 + k1;
#define TF_QR(r0, r1, r2, r3)                                \
    x0 += x1; x1 = rotl32(x1, r0); x1 ^= x0;                 \
    x0 += x1; x1 = rotl32(x1, r1); x1 ^= x0;                 \
    x0 += x1; x1 = rotl32(x1, r2); x1 ^= x0;                 \
    x0 += x1; x1 = rotl32(x1, r3); x1 ^= x0;
    TF_QR(13u, 15u, 26u, 6u)  x0 += k1;  x1 += ks2 + 1u;
    TF_QR(17u, 29u, 16u, 24u) x0 += ks2; x1 += k0 + 2u;
    TF_QR(13u, 15u, 26u, 6u)  x0 += k0;  x1 += k1 + 3u;
    TF_QR(17u, 29u, 16u, 24u) x0 += k1;  x1 += ks2 + 4u;
    TF_QR(13u, 15u, 26u, 6u)  x0 += ks2; x1 += k0 + 5u;
#undef TF_QR
    o0 = x0; o1 = x1;
}

// Uniform [0,1) score for flat index f over an n = B_*L_ array, JAX semantics.
__device__ unsigned jax_uniform_bits(unsigned f) {
    const unsigned half = (B_ * L_) / 2;             // 32768
    unsigned p = (f < half) ? f : (f - half);
    unsigned o0, o1;
    threefry2x32(0u, 42u, p, p + half, o0, o1);
    unsigned bits = (f < half) ? o0 : o1;
    float s = __uint_as_float((bits >> 9) | 0x3f800000u) - 1.0f;
    return __float_as_uint(s);                       // monotone key, [0, 0x3F800000)
}

// ---------------------------------------------------------------------------
// Block-wide radix select: returns the k-th largest key in skey[0..n).
// ---------------------------------------------------------------------------
__device__ unsigned radix_select_kth(const unsigned* skey, int n, int k, unsigned* cnt) {
    unsigned prefix = 0;
    int kk = k;
    for (int bit = 31; bit >= 0; --bit) {
        unsigned bm = 1u << bit;
        unsigned high = ~((bm << 1) - 1u);           // bits above 'bit' (0 when bit==31)
        if (threadIdx.x == 0) *cnt = 0;
        __syncthreads();
        unsigned local = 0;
        for (int i = threadIdx.x; i < n; i += blockDim.x) {
            unsigned kx = skey[i];
            if ((kx & high) == prefix && (kx & bm)) local++;
        }
        if (local) atomicAdd(cnt, local);
        __syncthreads();
        unsigned c = *cnt;
        __syncthreads();
        if (c >= (unsigned)kk) prefix |= bm;
        else kk -= (int)c;
    }
    return prefix;
}

// Keep all keys > T; then the first (k - count_gt) keys == T in index order
// (matches jax.lax.top_k stable tie-breaking).
__device__ void mark_keep(const unsigned* skey, unsigned char* keep, int n, int k,
                          unsigned T, unsigned* cnt) {
    if (threadIdx.x == 0) *cnt = 0;
    __syncthreads();
    unsigned local = 0;
    for (int i = threadIdx.x; i < n; i += blockDim.x) {
        if (skey[i] > T) { keep[i] = 1; local++; }
    }
    if (local) atomicAdd(cnt, local);
    __syncthreads();
    int neq = k - (int)(*cnt);
    __syncthreads();
    if (threadIdx.x == 0) {
        for (int i = 0; i < n && neq > 0; ++i) {
            if (skey[i] == T) { keep[i] = 1; neq--; }
        }
    }
    __syncthreads();
}

// ---------------------------------------------------------------------------
// Kernel 2: one block per batch; top-k on norms, then random top-k on
// Threefry scores over the remaining tokens; writes keep mask.
// ---------------------------------------------------------------------------
__global__ __launch_bounds__(1024) void select_kernel(const float* __restrict__ norms,
                                                      int* __restrict__ mask) {
    __shared__ unsigned skey[L_];
    __shared__ unsigned char keep[L_];
    __shared__ unsigned cnt;
    const int b   = blockIdx.x;
    const int tid = threadIdx.x;
    const int nt  = blockDim.x;

    for (int i = tid; i < L_; i += nt) {
        skey[i] = __float_as_uint(norms[b * L_ + i]);  // >=0 -> bit-order == value-order
        keep[i] = 0;
    }
    __syncthreads();

    // Stage 1: top NTOP by norm.
    unsigned T1 = radix_select_kth(skey, L_, NTOP, &cnt);
    mark_keep(skey, keep, L_, NTOP, T1, &cnt);
    __syncthreads();

    // Stage 2: random NRAND from the remainder (score keys; kept tokens -> 0).
    for (int i = tid; i < L_; i += nt) {
        unsigned sb = jax_uniform_bits((unsigned)(b * L_ + i));
        skey[i] = keep[i] ? 0u : ((sb << 1) | 1u);     // sb < 2^30, so no overflow
    }
    __syncthreads();
    unsigned T2 = radix_select_kth(skey, L_, NRAND, &cnt);
    mark_keep(skey, keep, L_, NRAND, T2, &cnt);
    __syncthreads();

    for (int i = tid; i < L_; i += nt) mask[b * L_ + i] = keep[i];
}

// ---------------------------------------------------------------------------
// Kernel 3: out = keep ? x : 0 (coalesced float4 pass)
// ---------------------------------------------------------------------------
__global__ __launch_bounds__(256) void scatter_kernel(const float4* __restrict__ x4,
                                                      const int* __restrict__ mask,
                                                      float4* __restrict__ out4) {
    int i = blockIdx.x * blockDim.x + threadIdx.x;     // < 8388608
    int token = i >> 7;                                // C_/4 = 128 float4 per token
    float4 v = x4[i];
    if (!mask[token]) { v.x = 0.f; v.y = 0.f; v.z = 0.f; v.w = 0.f; }
    out4[i] = v;
}

extern "C" void kernel_launch(void* const* d_in, const int* in_sizes, int n_in,
                              void* d_out, int out_size, void* d_ws, size_t ws_size,
                              hipStream_t stream) {
    (void)in_sizes; (void)n_in; (void)out_size; (void)ws_size;
    const float* x = (const float*)d_in[0];
    float* out = (float*)d_out;
    float* norms = (float*)d_ws;                                   // B_*L_ floats
    int*   mask  = (int*)((char*)d_ws + (size_t)B_ * L_ * sizeof(float)); // B_*L_ ints

    norms_wmma_kernel<<<(B_ * L_) / 128, 256, 0, stream>>>(x, norms);
    select_kernel<<<B_, 1024, 0, stream>>>(norms, mask);
    scatter_kernel<<<(B_ * L_ * C_ / 4) / 256, 256, 0, stream>>>(
        (const float4*)x, mask, (float4*)out);
}